// SampledYatXML_51986284151006
// MI455X (gfx1250) — compile-verified
//
#include <hip/hip_runtime.h>
#include <math.h>

// Problem constants (match reference)
#define B_DIM        1024
#define L_DIM        128
#define K_DIM        16
#define D_DIM        512
#define NUM_LABELS_C 131072
#define NUM_NEG_C    2048
#define EPS_C        0.1f
#define CC_C         2.1f   // C = 2.0 + EPS

typedef __attribute__((ext_vector_type(2))) float v2f;
typedef __attribute__((ext_vector_type(8))) float v8f;

// ds_swizzle_b32 XOR-mode butterfly add: offset = (xor<<10) | (or=0)<<5 | (and=0x1f)
// One DS instruction per step, no lane-bounds VALU overhead (vs __shfl_xor).
#define SWZ_ADD(v, off)                                                        \
    (v) += __int_as_float(__builtin_amdgcn_ds_swizzle(__float_as_int(v), (off)))

#define SWZ_XOR1  0x041F
#define SWZ_XOR2  0x081F
#define SWZ_XOR4  0x101F
#define SWZ_XOR8  0x201F
#define SWZ_XOR16 0x401F

// ---------------------------------------------------------------------------
// Kernel 1: enc = (sum_l embed[idx]*mask) / max(sum mask,1); query = enc/||enc||
// One block per row b, 128 threads, float4 lanes over D=512.
// Also zeroes rowsum[b] for the negative-GEMM accumulation.
// ---------------------------------------------------------------------------
__global__ void encode_kernel(const int* __restrict__ indices,
                              const float* __restrict__ mask,
                              const float* __restrict__ embed,
                              float* __restrict__ query,
                              float* __restrict__ rowsum) {
    __shared__ int   s_idx[L_DIM];
    __shared__ float s_m[L_DIM];
    __shared__ float s_red[128];

    const int b = blockIdx.x;
    const int t = threadIdx.x;      // 0..127

    s_idx[t] = indices[b * L_DIM + t];
    s_m[t]   = mask[b * L_DIM + t];
    __syncthreads();

    // sum of mask over L
    s_red[t] = s_m[t];
    __syncthreads();
    for (int s = 64; s > 0; s >>= 1) {
        if (t < s) s_red[t] += s_red[t + s];
        __syncthreads();
    }
    const float denom = fmaxf(s_red[0], 1.0f);
    __syncthreads();

    // gather-accumulate: float4 per thread (coalesced 128-bit loads)
    const float4* ept = (const float4*)embed;   // row = idx*128 float4s
    float4 acc = make_float4(0.f, 0.f, 0.f, 0.f);
    for (int i = 0; i < L_DIM; ++i) {
        const float  m = s_m[i];
        const float4 e = ept[(size_t)s_idx[i] * (D_DIM / 4) + t];
        acc.x += e.x * m; acc.y += e.y * m;
        acc.z += e.z * m; acc.w += e.w * m;
    }
    const float inv_d = 1.0f / denom;
    acc.x *= inv_d; acc.y *= inv_d; acc.z *= inv_d; acc.w *= inv_d;

    // L2 norm over D
    s_red[t] = acc.x * acc.x + acc.y * acc.y + acc.z * acc.z + acc.w * acc.w;
    __syncthreads();
    for (int s = 64; s > 0; s >>= 1) {
        if (t < s) s_red[t] += s_red[t + s];
        __syncthreads();
    }
    const float inv_n = 1.0f / fmaxf(sqrtf(s_red[0]), 1e-4f);

    float4 qv;
    qv.x = acc.x * inv_n; qv.y = acc.y * inv_n;
    qv.z = acc.z * inv_n; qv.w = acc.w * inv_n;
    ((float4*)query)[(size_t)b * (D_DIM / 4) + t] = qv;

    if (t == 0) rowsum[b] = 0.0f;
}

// ---------------------------------------------------------------------------
// Kernel 2: Wn[n,:] = normalize(W_kernel[:, neg_indices[n]])  (2048 x 512)
// One block per negative index, 256 threads x 2 elements.
// ---------------------------------------------------------------------------
__global__ void wn_kernel(const int* __restrict__ neg_idx,
                          const float* __restrict__ Wk,
                          float* __restrict__ Wn) {
    __shared__ float s_red[256];
    const int n = blockIdx.x;
    const int t = threadIdx.x;
    const int j = neg_idx[n];

    const float w0 = Wk[(size_t)t * NUM_LABELS_C + j];
    const float w1 = Wk[(size_t)(t + 256) * NUM_LABELS_C + j];

    s_red[t] = w0 * w0 + w1 * w1;
    __syncthreads();
    for (int s = 128; s > 0; s >>= 1) {
        if (t < s) s_red[t] += s_red[t + s];
        __syncthreads();
    }
    const float inv = 1.0f / fmaxf(sqrtf(s_red[0]), 1e-4f);
    Wn[(size_t)n * D_DIM + t]       = w0 * inv;
    Wn[(size_t)n * D_DIM + t + 256] = w1 * inv;
}

// ---------------------------------------------------------------------------
// Kernel 3: positive scores. One wave per (b,k):
//   x   = dot(query[b], W[:,lab]) / max(||W[:,lab]||, 1e-4), clipped
//   pos = x^2 / max(C - 2x, EPS)
// Wave reduction via single-instruction ds_swizzle XOR butterflies.
// ---------------------------------------------------------------------------
__global__ void pos_kernel(const int* __restrict__ labels,
                           const float* __restrict__ query,
                           const float* __restrict__ Wk,
                           float* __restrict__ pos_scores) {
    const int wid  = (blockIdx.x * blockDim.x + threadIdx.x) >> 5;
    const int lane = threadIdx.x & 31;
    const int b = wid >> 4;        // K = 16
    const int k = wid & 15;

    int idx = labels[b * K_DIM + k];
    if (idx < 0) idx = 0;

    float dot = 0.f, ssq = 0.f;
    for (int d = lane; d < D_DIM; d += 32) {
        const float qv = query[(size_t)b * D_DIM + d];
        const float wv = Wk[(size_t)d * NUM_LABELS_C + idx];
        dot += qv * wv;
        ssq += wv * wv;
    }
    SWZ_ADD(dot, SWZ_XOR1);  SWZ_ADD(ssq, SWZ_XOR1);
    SWZ_ADD(dot, SWZ_XOR2);  SWZ_ADD(ssq, SWZ_XOR2);
    SWZ_ADD(dot, SWZ_XOR4);  SWZ_ADD(ssq, SWZ_XOR4);
    SWZ_ADD(dot, SWZ_XOR8);  SWZ_ADD(ssq, SWZ_XOR8);
    SWZ_ADD(dot, SWZ_XOR16); SWZ_ADD(ssq, SWZ_XOR16);
    if (lane == 0) {
        float x = dot / fmaxf(sqrtf(ssq), 1e-4f);
        x = fminf(fmaxf(x, -0.999f), 0.999f);
        const float sc = x * x / fmaxf(CC_C - 2.0f * x, EPS_C);
        pos_scores[b * K_DIM + k] = sc;
    }
}

// ---------------------------------------------------------------------------
// Kernel 4: negative-score GEMM + transform + row-sum, via f32 WMMA.
//   dot = query (1024x512) @ Wn^T (512x2048), tile = 16x16 per wave,
//   K-loop of V_WMMA_F32_16X16X4_F32.
// A layout (16x4 f32): lanes 0-15 -> M=lane, K=0,1 in vgpr0,1;
//                      lanes 16-31 -> M=lane-16, K=2,3.
// B layout (4x16 f32): lanes 0-15 -> N=lane, K=0,1; lanes 16-31 -> N, K=2,3.
// C layout (16x16 f32): vgpr r, lanes 0-15 -> (M=r, N=lane);
//                       lanes 16-31 -> (M=r+8, N=lane-16).
// ---------------------------------------------------------------------------
__global__ void neg_gemm_kernel(const float* __restrict__ q,
                                const float* __restrict__ Wn,
                                float* __restrict__ rowsum) {
    const int lane = threadIdx.x & 31;
    const int wave = threadIdx.x >> 5;
    const int l    = lane & 15;
    const int half = lane >> 4;

    const int m0 = blockIdx.x * 16;                 // query rows
    const int n0 = (blockIdx.y * 8 + wave) * 16;    // negative rows

    const float* qrow = q  + (size_t)(m0 + l) * D_DIM + 2 * half;
    const float* wrow = Wn + (size_t)(n0 + l) * D_DIM + 2 * half;

    v8f c = {0.f, 0.f, 0.f, 0.f, 0.f, 0.f, 0.f, 0.f};

#pragma unroll 8
    for (int k = 0; k < D_DIM; k += 4) {
        v2f a  = *(const v2f*)(qrow + k);   // A[m, k + 2*half .. +1]
        v2f bb = *(const v2f*)(wrow + k);   // B[k + 2*half .. +1, n]
        c = __builtin_amdgcn_wmma_f32_16x16x4_f32(
                /*neg_a=*/false, a, /*neg_b=*/false, bb,
                /*c_mod=*/(short)0, c, /*reuse_a=*/false, /*reuse_b=*/false);
    }

    // score transform + reduce over N (16 lanes per half) with ds_swizzle
    // butterflies (stay within each 16-lane half since xor < 16), then one
    // atomic per output row from lanes 0 and 16.
#pragma unroll
    for (int r = 0; r < 8; ++r) {
        float x = c[r];
        x = fminf(fmaxf(x, -0.999f), 0.999f);
        float v = x * x / fmaxf(CC_C - 2.0f * x, EPS_C);
        SWZ_ADD(v, SWZ_XOR1);
        SWZ_ADD(v, SWZ_XOR2);
        SWZ_ADD(v, SWZ_XOR4);
        SWZ_ADD(v, SWZ_XOR8);
        if (l == 0) atomicAdd(&rowsum[m0 + r + 8 * half], v);
    }
}

// ---------------------------------------------------------------------------
// Kernel 5: loss = -sum_{b,k} (log(pos+1e-8) - log(total_b)) * lm / (sum lm + 1e-6)
// total_b = sum_k pos + rowsum_neg[b] + 2064e-8   (logsumexp(log x) == log(sum x))
// ---------------------------------------------------------------------------
__global__ void final_kernel(const float* __restrict__ pos_scores,
                             const float* __restrict__ rowsum,
                             const float* __restrict__ label_mask,
                             float* __restrict__ out) {
    __shared__ float s_num[1024];
    __shared__ float s_den[1024];
    const int b = threadIdx.x;

    float ps[K_DIM];
    float pos_sum = 0.f;
#pragma unroll
    for (int k = 0; k < K_DIM; ++k) {
        ps[k] = pos_scores[b * K_DIM + k];
        pos_sum += ps[k];
    }
    const float total = pos_sum + rowsum[b] + 2064.0f * 1e-8f;
    const float logZ  = logf(fmaxf(total, 1e-8f));

    float num = 0.f, den = 0.f;
#pragma unroll
    for (int k = 0; k < K_DIM; ++k) {
        const float lm = label_mask[b * K_DIM + k];
        num += (logf(fmaxf(ps[k] + 1e-8f, 1e-8f)) - logZ) * lm;
        den += lm;
    }
    s_num[b] = num;
    s_den[b] = den;
    __syncthreads();
    for (int s = 512; s > 0; s >>= 1) {
        if (b < s) { s_num[b] += s_num[b + s]; s_den[b] += s_den[b + s]; }
        __syncthreads();
    }
    if (b == 0) out[0] = -s_num[0] / (s_den[0] + 1e-6f);
}

// ---------------------------------------------------------------------------
extern "C" void kernel_launch(void* const* d_in, const int* in_sizes, int n_in,
                              void* d_out, int out_size, void* d_ws, size_t ws_size,
                              hipStream_t stream) {
    const int*   indices     = (const int*)  d_in[0];
    const float* mask        = (const float*)d_in[1];
    const int*   labels      = (const int*)  d_in[2];
    const float* label_mask  = (const float*)d_in[3];
    const int*   neg_indices = (const int*)  d_in[4];
    const float* embed       = (const float*)d_in[5];
    const float* Wk          = (const float*)d_in[6];

    float* ws         = (float*)d_ws;
    float* q          = ws;                          // 1024*512   = 524288 f32
    float* Wn         = q  + (size_t)B_DIM * D_DIM;  // 2048*512   = 1048576 f32
    float* rowsum     = Wn + (size_t)NUM_NEG_C * D_DIM;  // 1024 f32
    float* pos_scores = rowsum + B_DIM;              // 16384 f32  (~6.1 MB total)

    encode_kernel<<<B_DIM, 128, 0, stream>>>(indices, mask, embed, q, rowsum);
    wn_kernel<<<NUM_NEG_C, 256, 0, stream>>>(neg_indices, Wk, Wn);
    pos_kernel<<<(B_DIM * K_DIM) / 8, 256, 0, stream>>>(labels, q, Wk, pos_scores);
    neg_gemm_kernel<<<dim3(B_DIM / 16, NUM_NEG_C / 128), 256, 0, stream>>>(q, Wn, rowsum);
    final_kernel<<<1, 1024, 0, stream>>>(pos_scores, rowsum, label_mask, (float*)d_out);
}